// ClassifyModel_70970039599212
// MI455X (gfx1250) — compile-verified
//
#include <hip/hip_runtime.h>
#include <math.h>

#define NNODES 50000
#define NEDGES 600000
#define NGRAPH 64
#define IN_DIM 128
#define HID    256
#define OUT_D  256
#define KMAX   256
#define KPAD(K) ((K) + 4)

typedef __attribute__((ext_vector_type(2))) float v2f;
typedef __attribute__((ext_vector_type(8))) float v8f;

// ---------------- degree kernels ----------------
__global__ void deg_kernel(const int* __restrict__ src, const int* __restrict__ dst,
                           float* __restrict__ odeg, float* __restrict__ ideg, int E) {
    int e = blockIdx.x * blockDim.x + threadIdx.x;
    if (e >= E) return;
    atomicAdd(&odeg[src[e]], 1.0f);
    atomicAdd(&ideg[dst[e]], 1.0f);
}

__global__ void invsqrt_kernel(float* __restrict__ odeg, float* __restrict__ ideg, int n) {
    int i = blockIdx.x * blockDim.x + threadIdx.x;
    if (i >= n) return;
    odeg[i] = 1.0f / sqrtf(fmaxf(odeg[i], 1.0f));
    ideg[i] = 1.0f / sqrtf(fmaxf(ideg[i], 1.0f));
}

// ---------------- edge scatter: m[dst] += h[src] * odeg_inv[src] ----------------
__global__ void scatter_kernel(const float* __restrict__ h, const int* __restrict__ src,
                               const int* __restrict__ dst, const float* __restrict__ odeg_inv,
                               float* __restrict__ m, int d, int E) {
    int idx = blockIdx.x * blockDim.x + threadIdx.x;
    int chunks = d >> 2;                       // float4 chunks per row
    int e = idx / chunks;
    if (e >= E) return;
    int c = (idx - e * chunks) << 2;
    int s = src[e], t = dst[e];
    float w = odeg_inv[s];
    const float4 hv = *(const float4*)(h + (size_t)s * d + c);
    float* mp = m + (size_t)t * d + c;
    atomicAdd(mp + 0, hv.x * w);
    atomicAdd(mp + 1, hv.y * w);
    atomicAdd(mp + 2, hv.z * w);
    atomicAdd(mp + 3, hv.w * w);
}

// -------- async global -> LDS copy helpers (CDNA5 ASYNCcnt path) --------
__device__ __forceinline__ void async_g2l_b32(unsigned lds_off, const void* gaddr) {
    asm volatile("global_load_async_to_lds_b32 %0, %1, off"
                 :: "v"(lds_off), "v"((unsigned long long)(size_t)gaddr)
                 : "memory");
}
__device__ __forceinline__ void wait_async0() {
    asm volatile("s_wait_asynccnt 0x0" ::: "memory");
}

// ---------------- WMMA fp32 GEMM: Out = act((rowscale*A) @ W + bias) with optional BN ----
// A: MxK row-major, W: KxNout row-major. M%16==0, K%4==0, Nout%32==0, K<=256.
// block = 256 threads (8 waves). Block computes 16 rows x 256 cols; each wave owns two
// 16x16 N-tiles. The 16xK A tile is staged once into LDS via async global->LDS copies.
__global__ __launch_bounds__(256)
void wmma_gemm_kernel(const float* __restrict__ A, const float* __restrict__ W,
                      const float* __restrict__ bias, const float* __restrict__ rowscale,
                      const float* __restrict__ bn_g, const float* __restrict__ bn_b,
                      float bn_s, int relu,
                      float* __restrict__ Out, int M, int K, int Nout) {
    __shared__ float atile[16 * KPAD(KMAX)];
    const int tid  = threadIdx.x;
    const int wave = tid >> 5;
    const int lane = tid & 31;
    const int mtile = blockIdx.x;
    const int kp = KPAD(K);

    // ---- stage A tile (16 x K) into LDS (async, padded row stride) ----
    const float* Abase = A + (size_t)mtile * 16 * K;
#pragma unroll
    for (int r = 0; r < 16; ++r) {
        for (int k = tid; k < K; k += 256) {
            unsigned loff = (unsigned)(size_t)&atile[r * kp + k];
            async_g2l_b32(loff, Abase + (size_t)r * K + k);
        }
    }
    wait_async0();
    __syncthreads();

    const int ntbase = (blockIdx.y * 8 + wave) * 2;    // first of two N-tiles for this wave
    if (ntbase * 16 < Nout) {                           // wave-uniform
        const int r16 = lane & 15;
        const int kb  = (lane >> 4) * 2;               // 0 or 2
        const int row = mtile * 16 + r16;
        const float rs = rowscale ? rowscale[row] : 1.0f;
        const int col0 = ntbase * 16 + r16;
        const int col1 = col0 + 16;

        v8f acc0 = {}, acc1 = {};
        for (int k0 = 0; k0 < K; k0 += 4) {
            const float2 av = *(const float2*)&atile[r16 * kp + k0 + kb]; // ds_load_b64
            v2f a; a.x = av.x * rs; a.y = av.y * rs;
            const float* __restrict__ Wr = W + (size_t)(k0 + kb) * Nout;
            v2f b0, b1;
            b0.x = Wr[col0]; b0.y = Wr[Nout + col0];
            b1.x = Wr[col1]; b1.y = Wr[Nout + col1];
            acc0 = __builtin_amdgcn_wmma_f32_16x16x4_f32(false, a, false, b0,
                                                         (short)0, acc0, false, false);
            acc1 = __builtin_amdgcn_wmma_f32_16x16x4_f32(false, a, false, b1,
                                                         (short)0, acc1, false, false);
        }

        const float bs0 = bias ? bias[col0] : 0.0f;
        const float bs1 = bias ? bias[col1] : 0.0f;
        const float g0  = bn_g ? bn_g[col0] * bn_s : 1.0f;
        const float g1  = bn_g ? bn_g[col1] * bn_s : 1.0f;
        const float bb0 = bn_b ? bn_b[col0] : 0.0f;
        const float bb1 = bn_b ? bn_b[col1] : 0.0f;
        const int mbase = mtile * 16 + (lane >> 4) * 8;
#pragma unroll
        for (int v = 0; v < 8; ++v) {
            float v0 = (acc0[v] + bs0) * g0 + bb0;
            float v1 = (acc1[v] + bs1) * g1 + bb1;
            if (relu) { v0 = fmaxf(v0, 0.0f); v1 = fmaxf(v1, 0.0f); }
            Out[(size_t)(mbase + v) * Nout + col0] = v0;
            Out[(size_t)(mbase + v) * Nout + col1] = v1;
        }
    }
}

// ---------------- gate scores: gate[i] = h3[i,:] . gate_w + gate_b ----------------
__global__ void gate_kernel(const float* __restrict__ h3, const float* __restrict__ gw,
                            const float* __restrict__ gb, float* __restrict__ gate,
                            int n, int d) {
    int i = blockIdx.x * blockDim.x + threadIdx.x;
    if (i >= n) return;
    const float* row = h3 + (size_t)i * d;
    float acc = 0.0f;
    for (int f = 0; f < d; ++f) acc += row[f] * gw[f];
    gate[i] = acc + gb[0];
}

// ---------------- per-graph softmax stats (block per graph) ----------------
__global__ void stats_kernel(const float* __restrict__ gate, const int* __restrict__ gid,
                             float* __restrict__ gmax, float* __restrict__ dsum, int n) {
    const int b = blockIdx.x;
    __shared__ float red[256];
    float mx = -INFINITY;
    for (int i = threadIdx.x; i < n; i += blockDim.x)
        if (gid[i] == b) mx = fmaxf(mx, gate[i]);
    red[threadIdx.x] = mx; __syncthreads();
    for (int s = 128; s > 0; s >>= 1) {
        if (threadIdx.x < s) red[threadIdx.x] = fmaxf(red[threadIdx.x], red[threadIdx.x + s]);
        __syncthreads();
    }
    const float gm = red[0]; __syncthreads();
    float sm = 0.0f;
    for (int i = threadIdx.x; i < n; i += blockDim.x)
        if (gid[i] == b) sm += expf(gate[i] - gm);
    red[threadIdx.x] = sm; __syncthreads();
    for (int s = 128; s > 0; s >>= 1) {
        if (threadIdx.x < s) red[threadIdx.x] += red[threadIdx.x + s];
        __syncthreads();
    }
    if (threadIdx.x == 0) { gmax[b] = gm; dsum[b] = red[0]; }
}

// ---------------- per-node attention weight ----------------
__global__ void alpha_kernel(const float* __restrict__ gate, const int* __restrict__ gid,
                             const float* __restrict__ gmax, const float* __restrict__ dsum,
                             float* __restrict__ alpha, int n) {
    int i = blockIdx.x * blockDim.x + threadIdx.x;
    if (i >= n) return;
    int g = gid[i];
    float ds = dsum[g];
    alpha[i] = (ds > 0.0f) ? (expf(gate[i] - gmax[g]) / ds) : 0.0f;
}

// ---------------- attention pooling via atomics (16K accumulators in L2) -------------
__global__ void pool_kernel(const float* __restrict__ h3, const float* __restrict__ alpha,
                            const int* __restrict__ gid, float* __restrict__ pooled,
                            int n, int d) {
    int idx = blockIdx.x * blockDim.x + threadIdx.x;
    int chunks = d >> 2;
    int i = idx / chunks;
    if (i >= n) return;
    int c = (idx - i * chunks) << 2;
    int g = gid[i];
    float a = alpha[i];
    const float4 hv = *(const float4*)(h3 + (size_t)i * d + c);
    float* pp = pooled + (size_t)g * d + c;
    atomicAdd(pp + 0, hv.x * a);
    atomicAdd(pp + 1, hv.y * a);
    atomicAdd(pp + 2, hv.z * a);
    atomicAdd(pp + 3, hv.w * a);
}

// ---------------- final 64x64 @ 64x2 ----------------
__global__ void final_kernel(const float* __restrict__ z2, const float* __restrict__ w,
                             const float* __restrict__ b, float* __restrict__ out) {
    int idx = threadIdx.x;                     // 128 threads
    int i = idx >> 1, j = idx & 1;
    float acc = b[j];
    for (int k = 0; k < 64; ++k) acc += z2[i * 64 + k] * w[k * 2 + j];
    out[idx] = acc;
}

extern "C" void kernel_launch(void* const* d_in, const int* in_sizes, int n_in,
                              void* d_out, int out_size, void* d_ws, size_t ws_size,
                              hipStream_t stream) {
    const float* x       = (const float*)d_in[0];
    const int*   src     = (const int*)  d_in[1];
    const int*   dst     = (const int*)  d_in[2];
    const int*   gid     = (const int*)  d_in[3];
    const float* W0 = (const float*)d_in[4];  const float* b0 = (const float*)d_in[5];
    const float* W1 = (const float*)d_in[6];  const float* b1 = (const float*)d_in[7];
    const float* W2 = (const float*)d_in[8];  const float* b2 = (const float*)d_in[9];
    const float* gw = (const float*)d_in[10]; const float* gb = (const float*)d_in[11];
    const float* m1w = (const float*)d_in[12]; const float* m1b = (const float*)d_in[13];
    const float* bn1g = (const float*)d_in[14]; const float* bn1b = (const float*)d_in[15];
    const float* m2w = (const float*)d_in[16]; const float* m2b = (const float*)d_in[17];
    const float* bn2g = (const float*)d_in[18]; const float* bn2b = (const float*)d_in[19];
    const float* m3w = (const float*)d_in[20]; const float* m3b = (const float*)d_in[21];

    // workspace layout (floats)
    float* ws   = (float*)d_ws;
    float* odeg = ws;                              // N
    float* ideg = odeg + NNODES;                   // N
    float* m    = ideg + NNODES;                   // N*256
    float* h    = m + (size_t)NNODES * 256;        // N*256
    float* gate = h + (size_t)NNODES * 256;        // N
    float* alpha = gate + NNODES;                  // N
    float* gmax = alpha + NNODES;                  // 64
    float* dsum = gmax + NGRAPH;                   // 64
    float* pooled = dsum + NGRAPH;                 // 64*256
    float* z1   = pooled + NGRAPH * OUT_D;         // 64*128
    float* z2   = z1 + NGRAPH * 128;               // 64*64

    const float bn_s = 1.0f / sqrtf(1.0f + 1e-5f);

    // degrees
    hipMemsetAsync(odeg, 0, 2 * (size_t)NNODES * sizeof(float), stream);
    deg_kernel<<<(NEDGES + 255) / 256, 256, 0, stream>>>(src, dst, odeg, ideg, NEDGES);
    invsqrt_kernel<<<(NNODES + 255) / 256, 256, 0, stream>>>(odeg, ideg, NNODES);

    // ---- layer 1: aggregate x (d=128), GEMM -> h [N,256], relu ----
    hipMemsetAsync(m, 0, (size_t)NNODES * IN_DIM * sizeof(float), stream);
    {
        long threads = (long)NEDGES * (IN_DIM / 4);
        scatter_kernel<<<(threads + 255) / 256, 256, 0, stream>>>(x, src, dst, odeg, m, IN_DIM, NEDGES);
    }
    wmma_gemm_kernel<<<dim3(NNODES / 16, 1), 256, 0, stream>>>(
        m, W0, b0, ideg, nullptr, nullptr, 1.0f, 1, h, NNODES, IN_DIM, HID);

    // ---- layer 2: aggregate h (d=256), GEMM -> h, relu ----
    hipMemsetAsync(m, 0, (size_t)NNODES * HID * sizeof(float), stream);
    {
        long threads = (long)NEDGES * (HID / 4);
        scatter_kernel<<<(threads + 255) / 256, 256, 0, stream>>>(h, src, dst, odeg, m, HID, NEDGES);
    }
    wmma_gemm_kernel<<<dim3(NNODES / 16, 1), 256, 0, stream>>>(
        m, W1, b1, ideg, nullptr, nullptr, 1.0f, 1, h, NNODES, HID, HID);

    // ---- layer 3: aggregate h, GEMM -> h (no relu) ----
    hipMemsetAsync(m, 0, (size_t)NNODES * HID * sizeof(float), stream);
    {
        long threads = (long)NEDGES * (HID / 4);
        scatter_kernel<<<(threads + 255) / 256, 256, 0, stream>>>(h, src, dst, odeg, m, HID, NEDGES);
    }
    wmma_gemm_kernel<<<dim3(NNODES / 16, 1), 256, 0, stream>>>(
        m, W2, b2, ideg, nullptr, nullptr, 1.0f, 0, h, NNODES, HID, OUT_D);

    // ---- attention pooling ----
    gate_kernel<<<(NNODES + 255) / 256, 256, 0, stream>>>(h, gw, gb, gate, NNODES, OUT_D);
    stats_kernel<<<NGRAPH, 256, 0, stream>>>(gate, gid, gmax, dsum, NNODES);
    alpha_kernel<<<(NNODES + 255) / 256, 256, 0, stream>>>(gate, gid, gmax, dsum, alpha, NNODES);
    hipMemsetAsync(pooled, 0, (size_t)NGRAPH * OUT_D * sizeof(float), stream);
    {
        long threads = (long)NNODES * (OUT_D / 4);
        pool_kernel<<<(threads + 255) / 256, 256, 0, stream>>>(h, alpha, gid, pooled, NNODES, OUT_D);
    }

    // ---- MLP: 256 -> 128 -> 64 -> 2 (BN folded into epilogue) ----
    wmma_gemm_kernel<<<dim3(NGRAPH / 16, 1), 256, 0, stream>>>(
        pooled, m1w, m1b, nullptr, bn1g, bn1b, bn_s, 1, z1, NGRAPH, OUT_D, 128);
    wmma_gemm_kernel<<<dim3(NGRAPH / 16, 1), 256, 0, stream>>>(
        z1, m2w, m2b, nullptr, bn2g, bn2b, bn_s, 1, z2, NGRAPH, 128, 64);
    final_kernel<<<1, 128, 0, stream>>>(z2, m3w, m3b, (float*)d_out);
}